// TransGEN_70540542870037
// MI455X (gfx1250) — compile-verified
//
#include <hip/hip_runtime.h>

// ---------------------------------------------------------------------------
// TransGEN (basis-decomposed relational conv) for MI455X / gfx1250.
//
// Key optimizations (see analysis):
//  * dead-output elimination: only the 500 unseen rows are ever returned,
//    so only edges with dst in the unseen set are processed (~1% of edges).
//  * att x basis folded into per-edge-type weights W[t] (400 types), stored
//    transposed+padded [112][200] f32 so WMMA B fragments are contiguous
//    8-byte loads. 72 MB for both branches -> resident in the 192 MB L2.
//  * surviving edges bucketed by type; each bucket processed as dense
//    16-edge x [200x100] GEMM tiles with V_WMMA_F32_16X16X4_F32 (wave32),
//    A staged in LDS (stride 202 floats -> conflict-free fragment reads).
//  * 2 waves / workgroup: wave0 = mu branch, wave1 = sigma branch, sharing
//    the LDS A tile.  f32 accumulate, scatter-add via global_atomic_add_f32.
// ---------------------------------------------------------------------------

typedef float v2f __attribute__((ext_vector_type(2)));
typedef float v8f __attribute__((ext_vector_type(8)));

constexpr int N_NODES = 50000;
constexpr int N_REL   = 200;
constexpr int DE      = 100;          // entity dim (output dim)
constexpr int DIN     = 200;          // De + Dr (GEMM K)
constexpr int NB      = 16;           // bases
constexpr int NT      = 2 * N_REL;    // 400 edge types
constexpr int E2      = 600000;
constexpr int U       = 500;
constexpr int NPAD    = 112;          // DE padded to 7 * 16 (WMMA N tiles)
constexpr int KSTEPS  = DIN / 4;      // 50 WMMA K steps (16x16x4 f32)
constexpr int NTILES  = NPAD / 16;    // 7
constexpr int ASTRIDE = 202;          // LDS row stride (floats), bank-conflict free

// ---------------------------------------------------------------------------
__global__ void k_init(int* map, int* typeCount, int* typeCursor, int* cnt,
                       float* agg_mu, float* agg_sig) {
  int i = blockIdx.x * blockDim.x + threadIdx.x;
  if (i < N_NODES) map[i] = -1;
  if (i < NT) { typeCount[i] = 0; typeCursor[i] = 0; }
  if (i < U) cnt[i] = 0;
  if (i < U * DE) { agg_mu[i] = 0.f; agg_sig[i] = 0.f; }
}

// node -> slot map; atomicMax so the highest duplicate slot wins
// (matches "last scatter update wins" for x.at[unseen_index].set).
__global__ void k_map(const int* __restrict__ unseen_index, int* map) {
  int u = blockIdx.x * blockDim.x + threadIdx.x;
  if (u < U) atomicMax(&map[unseen_index[u]], u);
}

// Pass 1 over edges: per-type survivor counts + per-slot in-degree
// (reference mean divides by the FULL in-degree of the node).
__global__ void k_count(const int* __restrict__ edge_index,
                        const int* __restrict__ edge_type,
                        const int* __restrict__ map,
                        int* typeCount, int* cnt) {
  int e = blockIdx.x * blockDim.x + threadIdx.x;
  if (e >= E2) return;
  int slot = map[edge_index[E2 + e]];       // dst row of edge_index
  if (slot >= 0) {
    atomicAdd(&typeCount[edge_type[e]], 1);
    atomicAdd(&cnt[slot], 1);
  }
}

// Exclusive prefix over 400 type counts (trivial size -> single thread).
__global__ void k_scan(const int* __restrict__ typeCount, int* typeStart) {
  if (blockIdx.x == 0 && threadIdx.x == 0) {
    int acc = 0;
    for (int t = 0; t < NT; ++t) { typeStart[t] = acc; acc += typeCount[t]; }
  }
}

// Pass 2: compact surviving edges into per-type buckets.
__global__ void k_scatter(const int* __restrict__ edge_index,
                          const int* __restrict__ edge_type,
                          const int* __restrict__ rel_index,
                          const int* __restrict__ map,
                          const int* __restrict__ typeStart, int* typeCursor,
                          int* sel_src, int* sel_rel, int* sel_slot) {
  int e = blockIdx.x * blockDim.x + threadIdx.x;
  if (e >= E2) return;
  int slot = map[edge_index[E2 + e]];
  if (slot < 0) return;
  int t = edge_type[e];
  int pos = typeStart[t] + atomicAdd(&typeCursor[t], 1);
  sel_src[pos]  = edge_index[e];            // src row of edge_index
  sel_rel[pos]  = rel_index[e];
  sel_slot[pos] = slot;
}

// W[t] = sum_b att[t,b] * basis[b], stored TRANSPOSED + PADDED:
// Wt[t][n][k], n in [0,112) (rows >= 100 are zero), k in [0,200).
// This makes WMMA B fragments ((K,K+1) pair at fixed N) contiguous 8B loads.
__global__ void k_weights(const float* __restrict__ att,
                          const float* __restrict__ basis,
                          float* __restrict__ Wt) {
  int t = blockIdx.x;
  __shared__ float satt[NB];
  if (threadIdx.x < NB) satt[threadIdx.x] = att[t * NB + threadIdx.x];
  __syncthreads();
  float* w = Wt + (size_t)t * NPAD * DIN;
  for (int i = threadIdx.x; i < NPAD * DIN; i += blockDim.x) {
    int n = i / DIN, k = i - n * DIN;
    float v = 0.f;
    if (n < DE) {
#pragma unroll
      for (int b = 0; b < NB; ++b)
        v = fmaf(satt[b], basis[((size_t)b * DIN + k) * DE + n], v);
    }
    w[i] = v;
  }
}

// ---------------------------------------------------------------------------
// WMMA message GEMM.  One block per edge type; 64 threads = 2 waves.
// wave0 -> mu branch, wave1 -> sigma branch (same A tile, different B).
// A tile: 16 edges x 200 (concat(x_src, rel_emb)) staged in LDS.
// 50 K-steps x 7 N-tiles of v_wmma_f32_16x16x4_f32 per tile per wave.
// ---------------------------------------------------------------------------
__global__ void k_gemm(const int* __restrict__ typeStart,
                       const int* __restrict__ typeCount,
                       const int* __restrict__ sel_src,
                       const int* __restrict__ sel_rel,
                       const int* __restrict__ sel_slot,
                       const float* __restrict__ Wmu,
                       const float* __restrict__ Wsig,
                       const float* __restrict__ ent,
                       const int* __restrict__ node_id,
                       const float* __restrict__ tue,
                       const int* __restrict__ map,
                       const float* __restrict__ rel_table,
                       float* agg_mu, float* agg_sig) {
  int t = blockIdx.x;
  int cntT = typeCount[t];
  if (cntT == 0) return;
  int start = typeStart[t];

  int wave = threadIdx.x >> 5;              // 0 = mu, 1 = sigma
  int lane = threadIdx.x & 31;
  const float* W = (wave ? Wsig : Wmu) + (size_t)t * NPAD * DIN;
  float* agg = wave ? agg_sig : agg_mu;

  // warm L2 with this block's 89.6 KB W panel (global_prefetch_b8)
  __builtin_prefetch(W + (size_t)threadIdx.x * (NPAD * DIN / 64), 0, 3);

  __shared__ __align__(16) float Am[16 * ASTRIDE];   // ~12.6 KB

  int rowA   = lane & 15;                   // A fragment: M = lane%16
  int kshift = (lane >> 4) * 2;             // lanes 16-31 hold K+2,K+3
  int ncol   = lane & 15;                   // B/C fragment: N = lane%16
  int rbase  = (lane >> 4) * 8;             // C/D: lanes 16-31 hold rows 8-15

  for (int base = 0; base < cntT; base += 16) {
    int nrows = min(16, cntT - base);
    __syncthreads();                        // protect prior-iteration reads

    // Cooperative LDS fill: 16 rows x 100 float2 (x_src | rel_emb), zero tail.
    for (int i = threadIdx.x; i < 16 * (DIN / 2); i += 64) {
      int r = i / (DIN / 2), c2 = i - r * (DIN / 2);   // c2 = float2 index
      float2 v = make_float2(0.f, 0.f);
      if (r < nrows) {
        int j = start + base + r;
        if (c2 < DE / 2) {                  // x_src half (cols 0..99)
          int s = sel_src[j];
          int ms = map[s];                  // unseen src -> overwritten row
          const float* xrow = (ms >= 0) ? (tue + (size_t)ms * DE)
                                        : (ent + (size_t)node_id[s] * DE);
          v = *(const float2*)(xrow + 2 * c2);
        } else {                            // rel_emb half (cols 100..199)
          v = *(const float2*)(rel_table + (size_t)sel_rel[j] * DE
                               + 2 * (c2 - DE / 2));
        }
      }
      *(float2*)(&Am[r * ASTRIDE + 2 * c2]) = v;   // col == 2*c2 both halves
    }
    __syncthreads();

    v8f acc[NTILES] = {};                   // 7 x 8 f32 accumulators
#pragma unroll 2
    for (int ks = 0; ks < KSTEPS; ++ks) {
      int kA = ks * 4 + kshift;
      v2f a = *(const v2f*)(&Am[rowA * ASTRIDE + kA]);   // A: M=rowA, K=kA,kA+1
#pragma unroll
      for (int nt = 0; nt < NTILES; ++nt) {
        // B: K=kA,kA+1 at N = nt*16+ncol  (contiguous in transposed W)
        v2f b = *(const v2f*)(W + (size_t)(nt * 16 + ncol) * DIN + kA);
        acc[nt] = __builtin_amdgcn_wmma_f32_16x16x4_f32(
            false, a, false, b, (short)0, acc[nt], false, false);
      }
    }

    // Scatter-add D into per-slot accumulators (f32 L2 atomics).
#pragma unroll
    for (int v = 0; v < 8; ++v) {
      int r = rbase + v;                    // edge row within tile
      if (r < nrows) {
        int slot = sel_slot[start + base + r];
        float* arow = agg + (size_t)slot * DE;
#pragma unroll
        for (int nt = 0; nt < NTILES; ++nt) {
          int col = nt * 16 + ncol;
          if (col < DE) atomicAdd(&arow[col], acc[nt][v]);
        }
      }
    }
  }
}

// mean divide + x@root + bias, gathered through the duplicate map into the
// three output sections (unseen == mu[unseen], then logvar[unseen]).
__global__ void k_out(const int* __restrict__ unseen_index,
                      const int* __restrict__ map,
                      const int* __restrict__ cnt,
                      const float* __restrict__ agg_mu,
                      const float* __restrict__ agg_sig,
                      const float* __restrict__ tue,
                      const float* __restrict__ root_mu,
                      const float* __restrict__ bias_mu,
                      const float* __restrict__ root_sg,
                      const float* __restrict__ bias_sg,
                      float* __restrict__ out) {
  int i = blockIdx.x * blockDim.x + threadIdx.x;
  if (i >= U * DE) return;
  int u = i / DE, col = i - u * DE;
  int w = map[unseen_index[u]];             // winning slot for this node
  const float* xrow = tue + (size_t)w * DE; // x at unseen node == tue[winner]
  float inv = 1.f / fmaxf((float)cnt[w], 1.f);
  float mu = agg_mu[(size_t)w * DE + col] * inv;
  float sg = agg_sig[(size_t)w * DE + col] * inv;
#pragma unroll 4
  for (int k = 0; k < DE; ++k) {
    float xv = xrow[k];
    mu = fmaf(xv, root_mu[k * DE + col], mu);
    sg = fmaf(xv, root_sg[k * DE + col], sg);
  }
  mu += bias_mu[col];
  sg += bias_sg[col];
  out[i] = mu;                              // unseen
  out[U * DE + i] = mu;                     // mu[unseen]
  out[2 * U * DE + i] = sg;                 // logvar[unseen]
}

// ---------------------------------------------------------------------------
extern "C" void kernel_launch(void* const* d_in, const int* in_sizes, int n_in,
                              void* d_out, int out_size, void* d_ws, size_t ws_size,
                              hipStream_t stream) {
  (void)in_sizes; (void)n_in; (void)out_size; (void)ws_size;
  const float* ent      = (const float*)d_in[0];
  const float* rel_tab  = (const float*)d_in[1];
  const float* tue      = (const float*)d_in[2];
  const float* att_mu   = (const float*)d_in[3];
  const float* basis_mu = (const float*)d_in[4];
  const float* root_mu  = (const float*)d_in[5];
  const float* bias_mu  = (const float*)d_in[6];
  const float* att_sg   = (const float*)d_in[7];
  const float* basis_sg = (const float*)d_in[8];
  const float* root_sg  = (const float*)d_in[9];
  const float* bias_sg  = (const float*)d_in[10];
  const int* node_id      = (const int*)d_in[11];
  const int* edge_index   = (const int*)d_in[12];
  const int* edge_type    = (const int*)d_in[13];
  const int* rel_index    = (const int*)d_in[14];
  const int* unseen_index = (const int*)d_in[15];
  float* out = (float*)d_out;

  // workspace carve (256B aligned): ~80 MB total
  char* ws = (char*)d_ws;
  size_t off = 0;
  auto carve = [&](size_t bytes) -> char* {
    char* p = ws + off;
    off = (off + bytes + 255) & ~(size_t)255;
    return p;
  };
  int*   map        = (int*)carve((size_t)N_NODES * 4);
  int*   typeCount  = (int*)carve((size_t)NT * 4);
  int*   typeStart  = (int*)carve((size_t)NT * 4);
  int*   typeCursor = (int*)carve((size_t)NT * 4);
  int*   cnt        = (int*)carve((size_t)U * 4);
  int*   sel_src    = (int*)carve((size_t)E2 * 4);
  int*   sel_rel    = (int*)carve((size_t)E2 * 4);
  int*   sel_slot   = (int*)carve((size_t)E2 * 4);
  float* Wmu        = (float*)carve((size_t)NT * NPAD * DIN * 4);  // 35.8 MB
  float* Wsig       = (float*)carve((size_t)NT * NPAD * DIN * 4);  // 35.8 MB
  float* agg_mu     = (float*)carve((size_t)U * DE * 4);
  float* agg_sig    = (float*)carve((size_t)U * DE * 4);

  dim3 b256(256);
  k_init<<<dim3((N_NODES + 255) / 256), b256, 0, stream>>>(
      map, typeCount, typeCursor, cnt, agg_mu, agg_sig);
  k_map<<<dim3((U + 255) / 256), b256, 0, stream>>>(unseen_index, map);
  // per-type weights can overlap the edge-filter pipeline on the stream
  k_weights<<<dim3(NT), b256, 0, stream>>>(att_mu, basis_mu, Wmu);
  k_weights<<<dim3(NT), b256, 0, stream>>>(att_sg, basis_sg, Wsig);
  k_count<<<dim3((E2 + 255) / 256), b256, 0, stream>>>(
      edge_index, edge_type, map, typeCount, cnt);
  k_scan<<<dim3(1), dim3(32), 0, stream>>>(typeCount, typeStart);
  k_scatter<<<dim3((E2 + 255) / 256), b256, 0, stream>>>(
      edge_index, edge_type, rel_index, map, typeStart, typeCursor,
      sel_src, sel_rel, sel_slot);
  k_gemm<<<dim3(NT), dim3(64), 0, stream>>>(
      typeStart, typeCount, sel_src, sel_rel, sel_slot, Wmu, Wsig,
      ent, node_id, tue, map, rel_tab, agg_mu, agg_sig);
  k_out<<<dim3((U * DE + 255) / 256), b256, 0, stream>>>(
      unseen_index, map, cnt, agg_mu, agg_sig, tue,
      root_mu, bias_mu, root_sg, bias_sg, out);
}